// RQBottleneck_27032524161608
// MI455X (gfx1250) — compile-verified
//
#include <hip/hip_runtime.h>
#include <hip/hip_bf16.h>
#include <stdint.h>
#include <math.h>

// ---------------------------------------------------------------------------
// RQ-VAE bottleneck (depth=4, K=4096, dim=128, N=65536) for gfx1250 / MI455X.
// Bandwidth-dominated: 1 GiB cost matrix per level, ~7.5 GiB traffic/level.
// GEMM via V_WMMA_F32_16X16X4_F32, tile staging via TDM tensor_load_to_lds.
// ---------------------------------------------------------------------------

typedef float v2f __attribute__((ext_vector_type(2)));
typedef float v8f __attribute__((ext_vector_type(8)));

#define N_PTS   65536
#define K_CODES 4096
#define EDIM    128
#define DEPTH   4
#define NKTOT   ((size_t)N_PTS * (size_t)K_CODES)

// ---- workspace layout (units: floats) -------------------------------------
#define WS_SCAL   ((size_t)0)   // [0]=sum [1]=sumsq [2]=minkey [3]=s_scale [4]=s_off [5]=esum [6]=g [7]=nsum
#define WS_XSQ    ((size_t)64)
#define WS_CSQ    (WS_XSQ + N_PTS)
#define WS_U      (WS_CSQ + K_CODES)
#define WS_LOGU   (WS_U + K_CODES)
#define WS_S      (WS_LOGU + K_CODES)
#define WS_CS     (WS_S + K_CODES)
#define WS_CSN    (WS_CS + K_CODES)
#define WS_V      (WS_CSN + K_CODES)
#define WS_IDX    (WS_V + N_PTS)
#define WS_VS     (WS_IDX + N_PTS)
#define WS_RES    (WS_VS + (size_t)K_CODES * EDIM)
#define WS_E      (WS_RES + (size_t)N_PTS * EDIM)   // bf16 E matrix: NKTOT/2 float slots
#define WS_D      (WS_E + NKTOT / 2)                // f32 d matrix: NKTOT float slots

// ---- output layout (units: floats) ----------------------------------------
#define OUT_Q     ((size_t)0)
#define OUT_CODES ((size_t)N_PTS * EDIM)                       // 8388608
#define OUT_NW    (OUT_CODES + (size_t)N_PTS * DEPTH)          // 8650752
#define OUT_NCS   (OUT_NW + (size_t)DEPTH * (K_CODES + 1) * EDIM)
#define OUT_NE    (OUT_NCS + (size_t)DEPTH * K_CODES)

#define TILE 128
#define LDSP 132   // +4 floats row pad => conflict-free ds_load_b64 fragments

// ===========================================================================
// TDM: 2D tile load Global -> LDS (D# groups built per CDNA5 ISA)
// ===========================================================================
#if __has_builtin(__builtin_amdgcn_tensor_load_to_lds)
#define HAVE_TDM 1
#else
#define HAVE_TDM 0
#endif

#if HAVE_TDM
typedef uint32_t tdm_v4u __attribute__((ext_vector_type(4)));
typedef int      tdm_v8i __attribute__((ext_vector_type(8)));
typedef int      tdm_v4i __attribute__((ext_vector_type(4)));

__device__ __forceinline__ void tdm_load_tile_f32(uint32_t lds_byte_off,
                                                  const float* gsrc,
                                                  uint32_t rows, uint32_t cols,
                                                  uint32_t row_stride_elems,
                                                  bool pad_rows) {
  uint64_t ga = (uint64_t)(uintptr_t)gsrc;
  tdm_v4u g0;
  g0[0] = 1u;                                   // count=1, user descriptor
  g0[1] = lds_byte_off;                         // lds_addr
  g0[2] = (uint32_t)(ga & 0xffffffffull);       // global_addr[31:0]
  g0[3] = (uint32_t)((ga >> 32) & 0x1ffffffull) | (2u << 30);  // addr[56:32] | type=2
  uint32_t big_d1 = 0x7fffffffu;                // generous tensor extent (tile always in-bounds)
  tdm_v8i g1;
  uint32_t w0 = (2u << 16);                     // data_size = 4 bytes
  if (pad_rows) w0 |= (1u << 20) | (6u << 22) | (3u << 25); // pad: every 128 dwords, +4 dwords
  g1[0] = (int)w0;
  g1[1] = (int)((cols & 0xffffu) << 16);                     // tensor_dim0[15:0]
  g1[2] = (int)(((cols >> 16) & 0xffffu) | ((big_d1 & 0xffffu) << 16));
  g1[3] = (int)(((big_d1 >> 16) & 0xffffu) | ((cols & 0xffffu) << 16)); // tile_dim0
  g1[4] = (int)(rows & 0xffffu);                             // tile_dim1 (tile_dim2=0)
  g1[5] = (int)(row_stride_elems);                           // tensor_dim0_stride[31:0]
  g1[6] = 0;                                                 // stride0 hi / stride1 lo
  g1[7] = 0;
  tdm_v4i z4 = {0, 0, 0, 0};
  tdm_v8i z8 = {0, 0, 0, 0, 0, 0, 0, 0};
  __builtin_amdgcn_tensor_load_to_lds(g0, g1, z4, z4, z8, 0);
}
#endif

// ===========================================================================
// small utility kernels
// ===========================================================================
__global__ void fill_f_kernel(float* p, float v, size_t n) {
  size_t i = (size_t)blockIdx.x * blockDim.x + threadIdx.x;
  size_t st = (size_t)gridDim.x * blockDim.x;
  for (; i < n; i += st) p[i] = v;
}
__global__ void fill_u_kernel(uint32_t* p, uint32_t v, size_t n) {
  size_t i = (size_t)blockIdx.x * blockDim.x + threadIdx.x;
  size_t st = (size_t)gridDim.x * blockDim.x;
  for (; i < n; i += st) p[i] = v;
}
__global__ void copy_f_kernel(const float* __restrict__ s, float* __restrict__ d, size_t n) {
  size_t i = (size_t)blockIdx.x * blockDim.x + threadIdx.x;
  size_t st = (size_t)gridDim.x * blockDim.x;
  for (; i < n; i += st) d[i] = s[i];
}

// per-row squared L2 norm of 128-dim rows; one wave32 per row
__global__ void rownorm_kernel(const float* __restrict__ src, float* __restrict__ out, int rows) {
  int row = blockIdx.x * 8 + (threadIdx.x >> 5);
  int lane = threadIdx.x & 31;
  if (row >= rows) return;
  float4 v = ((const float4*)(src + (size_t)row * EDIM))[lane];
  float s = v.x * v.x + v.y * v.y + v.z * v.z + v.w * v.w;
  for (int o = 16; o > 0; o >>= 1) s += __shfl_down(s, o);
  if (lane == 0) out[row] = s;
}

// ===========================================================================
// WMMA distance GEMM: d[b,k] = |x_b|^2 + |c_k|^2 - 2 x_b . c_k
// block = 512 threads (16 waves), 128x128 output tile, f32 WMMA 16x16x4.
// Also accumulates sum / sumsq / min of d for the normalization stats.
// ===========================================================================
__device__ __forceinline__ uint32_t f32_order_key(float f) {
  uint32_t b = __float_as_uint(f);
  return (b & 0x80000000u) ? ~b : (b | 0x80000000u);
}

__global__ __launch_bounds__(512) void gemm_dist_kernel(
    const float* __restrict__ X,    // residual, N_PTS x 128
    const float* __restrict__ CB,   // codebook, K_CODES x 128
    const float* __restrict__ xsq,
    const float* __restrict__ csq,
    float* __restrict__ Dm,         // N_PTS x K_CODES
    float* __restrict__ stats) {    // [0]=sum [1]=sumsq [2]=minkey
  extern __shared__ float lds[];
  float* shX = lds;                  // 128 x 132
  float* shC = lds + TILE * LDSP;    // 128 x 132
  const int m0 = blockIdx.y * TILE;
  const int n0 = blockIdx.x * TILE;

#if HAVE_TDM
  if (threadIdx.x == 0) {
    tdm_load_tile_f32((uint32_t)(uintptr_t)shX, X + (size_t)m0 * EDIM, TILE, EDIM, EDIM, true);
    tdm_load_tile_f32((uint32_t)(uintptr_t)shC, CB + (size_t)n0 * EDIM, TILE, EDIM, EDIM, true);
    __builtin_amdgcn_s_wait_tensorcnt(0);
  }
  __syncthreads();
#else
  for (int i = threadIdx.x; i < TILE * TILE / 4; i += 512) {
    int r = i >> 5, c4 = (i & 31) * 4;
    *(float4*)&shX[r * LDSP + c4] = *(const float4*)&X[(size_t)(m0 + r) * EDIM + c4];
    *(float4*)&shC[r * LDSP + c4] = *(const float4*)&CB[(size_t)(n0 + r) * EDIM + c4];
  }
  __syncthreads();
#endif

  const int w = threadIdx.x >> 5;       // wave 0..15
  const int lane = threadIdx.x & 31;
  const int wr = w >> 2, wc = w & 3;    // 4x4 waves, 32x32 region each
  const int h = lane >> 4, l15 = lane & 15;

  v8f acc[2][2] = {};
  const int ro0 = (wr * 32 + l15) * LDSP + 2 * h;       // A rows
  const int ro1 = (wr * 32 + 16 + l15) * LDSP + 2 * h;
  const int co0 = (wc * 32 + l15) * LDSP + 2 * h;       // B rows (codebook entries)
  const int co1 = (wc * 32 + 16 + l15) * LDSP + 2 * h;

  for (int k0 = 0; k0 < EDIM; k0 += 4) {
    v2f a0 = *(const v2f*)&shX[ro0 + k0];
    v2f a1 = *(const v2f*)&shX[ro1 + k0];
    v2f b0 = *(const v2f*)&shC[co0 + k0];
    v2f b1 = *(const v2f*)&shC[co1 + k0];
    acc[0][0] = __builtin_amdgcn_wmma_f32_16x16x4_f32(false, a0, false, b0, (short)0, acc[0][0], false, false);
    acc[0][1] = __builtin_amdgcn_wmma_f32_16x16x4_f32(false, a0, false, b1, (short)0, acc[0][1], false, false);
    acc[1][0] = __builtin_amdgcn_wmma_f32_16x16x4_f32(false, a1, false, b0, (short)0, acc[1][0], false, false);
    acc[1][1] = __builtin_amdgcn_wmma_f32_16x16x4_f32(false, a1, false, b1, (short)0, acc[1][1], false, false);
  }

  // epilogue: C/D layout — VGPR e: lanes0-15 -> M=e, lanes16-31 -> M=8+e; N=l15
  float lsum = 0.f, lsq = 0.f, lmin = 3.4e38f;
  for (int i = 0; i < 2; ++i)
    for (int j = 0; j < 2; ++j)
      for (int e = 0; e < 8; ++e) {
        int gm = m0 + wr * 32 + i * 16 + 8 * h + e;
        int gn = n0 + wc * 32 + j * 16 + l15;
        float dv = xsq[gm] + csq[gn] - 2.0f * acc[i][j][e];
        Dm[(size_t)gm * K_CODES + gn] = dv;
        lsum += dv; lsq += dv * dv; lmin = fminf(lmin, dv);
      }

  for (int o = 16; o > 0; o >>= 1) {
    lsum += __shfl_down(lsum, o);
    lsq  += __shfl_down(lsq, o);
    lmin = fminf(lmin, __shfl_down(lmin, o));
  }
  __syncthreads();               // tiles no longer needed; reuse shX as scratch
  if (lane == 0) { shX[w] = lsum; shX[32 + w] = lsq; shX[64 + w] = lmin; }
  __syncthreads();
  if (threadIdx.x == 0) {
    float bs = 0.f, bq = 0.f, bm = 3.4e38f;
    for (int i = 0; i < 16; ++i) { bs += shX[i]; bq += shX[32 + i]; bm = fminf(bm, shX[64 + i]); }
    atomicAdd(&stats[0], bs);
    atomicAdd(&stats[1], bq);
    atomicMin((uint32_t*)&stats[2], f32_order_key(bm));
  }
}

// dn = (d-mean)/std(ddof=1); dn -= min(dn)  =>  exp(-eps*dn) = exp(s*d + o)
__global__ void finalize_stats_kernel(float* F) {
  double M = (double)NKTOT;
  double sum = (double)F[0], sq = (double)F[1];
  double mean = sum / M;
  double var = (sq - sum * sum / M) / (M - 1.0);
  float stdv = sqrtf(fmaxf((float)var, 0.0f));
  float inv = 1.0f / (stdv + 1e-6f);
  uint32_t key = ((uint32_t*)F)[2];
  uint32_t bits = (key & 0x80000000u) ? (key ^ 0x80000000u) : ~key;
  float dmin = __uint_as_float(bits);
  F[3] = -10.0f * inv;          // s_scale  (SINKHORN_EPSILON = 10)
  F[4] = 10.0f * inv * dmin;    // s_off = -s_scale * dmin
}

// E = exp(s*d + o) cached as bf16; also reduce total sum -> F[5]
__global__ void build_E_kernel(const float* __restrict__ Dm, __hip_bfloat16* __restrict__ E,
                               const float* __restrict__ F, float* esum) {
  float s = F[3], o = F[4];
  size_t i = (size_t)blockIdx.x * blockDim.x + threadIdx.x;
  size_t st = (size_t)gridDim.x * blockDim.x;
  float acc = 0.f;
  for (; i < NKTOT; i += st) {
    float e = __expf(fmaf(s, Dm[i], o));
    E[i] = __float2bfloat16(e);
    acc += e;
  }
  for (int d = 16; d > 0; d >>= 1) acc += __shfl_down(acc, d);
  if ((threadIdx.x & 31) == 0) atomicAdd(esum, acc);
}

__global__ void finalize_g_kernel(float* F) { F[6] = 1.0f / (F[5] + 1e-8f); }

// S[k] = sum_b v[b] * E[b,k]   (256-row x 256-col tiles, coalesced bf16 reads)
__global__ __launch_bounds__(256) void rowsumS_kernel(const __hip_bfloat16* __restrict__ E,
                                                      const float* __restrict__ v,
                                                      float* __restrict__ S) {
  int c = blockIdx.x * 256 + threadIdx.x;
  int b0 = blockIdx.y * 256;
  float acc = 0.f;
  for (int r = 0; r < 256; ++r)
    acc += v[b0 + r] * __bfloat162float(E[(size_t)(b0 + r) * K_CODES + c]);
  atomicAdd(&S[c], acc);
}

// u <- u / (g*u*S + 1e-8) / K ; logu for the argmax
__global__ void update_u_kernel(float* __restrict__ u, float* __restrict__ logu,
                                const float* __restrict__ S, const float* __restrict__ F) {
  int k = blockIdx.x * 256 + threadIdx.x;
  if (k >= K_CODES) return;
  float g = F[6];
  float uk = u[k];
  float r = g * uk * S[k];
  float un = uk / (r + 1e-8f) * (1.0f / (float)K_CODES);
  u[k] = un;
  logu[k] = __logf(un);
}

// per-row b: T = sum_k u[k]*E[b,k] ; v <- v/(g*v*T + 1e-8)/B
__global__ __launch_bounds__(256) void colsum_v_kernel(const __hip_bfloat16* __restrict__ E,
                                                       const float* __restrict__ u,
                                                       float* __restrict__ v,
                                                       const float* __restrict__ F) {
  __shared__ float red[8];
  int b = blockIdx.x, t = threadIdx.x;
  const __hip_bfloat16* row = E + (size_t)b * K_CODES;
  float acc = 0.f;
  for (int c = t; c < K_CODES; c += 256) acc += u[c] * __bfloat162float(row[c]);
  for (int o = 16; o > 0; o >>= 1) acc += __shfl_down(acc, o);
  if ((t & 31) == 0) red[t >> 5] = acc;
  __syncthreads();
  if (t == 0) {
    float T = 0.f;
    for (int i = 0; i < 8; ++i) T += red[i];
    float g = F[6], vb = v[b];
    float cb = g * vb * T;
    v[b] = vb / (cb + 1e-8f) * (1.0f / (float)N_PTS);
  }
}

// idx[b] = argmax_k (logu[k] + s_scale*d[b,k]) — row factors drop out of argmax
__global__ __launch_bounds__(256) void argmax_kernel(const float* __restrict__ Dm,
                                                     const float* __restrict__ logu,
                                                     const float* __restrict__ F,
                                                     int* __restrict__ idx,
                                                     float* __restrict__ codes_out, int lvl) {
  __shared__ float rs[256];
  __shared__ int ri[256];
  int b = blockIdx.x, t = threadIdx.x;
  float s = F[3];
  const float* row = Dm + (size_t)b * K_CODES;
  float best = -3.4e38f; int bi = 0;
  for (int c = t; c < K_CODES; c += 256) {
    float sc = logu[c] + s * row[c];
    if (sc > best) { best = sc; bi = c; }
  }
  rs[t] = best; ri[t] = bi;
  __syncthreads();
  for (int o = 128; o > 0; o >>= 1) {
    if (t < o) {
      if (rs[t + o] > rs[t] || (rs[t + o] == rs[t] && ri[t + o] < ri[t])) {
        rs[t] = rs[t + o]; ri[t] = ri[t + o];
      }
    }
    __syncthreads();
  }
  if (t == 0) {
    idx[b] = ri[0];
    codes_out[(size_t)b * DEPTH + lvl] = (float)ri[0];
  }
}

// segment sums: cs[k] += 1, vs[k,:] += x[b,:]
__global__ void seg_kernel(const int* __restrict__ idx, const float* __restrict__ res,
                           float* __restrict__ cs, float* __restrict__ vs) {
  int b = blockIdx.x, t = threadIdx.x;   // 128 threads
  int k = idx[b];
  atomicAdd(&vs[(size_t)k * EDIM + t], res[(size_t)b * EDIM + t]);
  if (t == 0) atomicAdd(&cs[k], 1.0f);
}

__global__ void ema_cs_kernel(const float* __restrict__ cse, const float* __restrict__ cs,
                              float* __restrict__ csn, float* __restrict__ ncs_out,
                              float* nsum) {
  int k = blockIdx.x * 256 + threadIdx.x;
  if (k >= K_CODES) return;
  float c = cse[k] * 0.99f + 0.01f * cs[k];
  csn[k] = c;
  ncs_out[k] = c;
  atomicAdd(nsum, c);
}

__global__ void upW_kernel(const float* __restrict__ ee, const float* __restrict__ vs,
                           const float* __restrict__ csn, const float* __restrict__ F,
                           float* __restrict__ ne_out, float* __restrict__ nw_out) {
  int i = blockIdx.x * 256 + threadIdx.x;
  if (i >= K_CODES * EDIM) return;
  int k = i >> 7;
  float n = F[7];
  float e = ee[i] * 0.99f + 0.01f * vs[i];
  ne_out[i] = e;
  float norm = n * (csn[k] + 1e-5f) / (n + (float)K_CODES * 1e-5f);
  nw_out[i] = e / norm;
}

__global__ void lastrow_kernel(const float* __restrict__ W, float* __restrict__ nw_out) {
  int t = threadIdx.x;  // 128
  nw_out[(size_t)K_CODES * EDIM + t] = W[(size_t)K_CODES * EDIM + t];
}

// residual -= W[idx]; quant (+)= W[idx]  (quant_st == quant numerically)
__global__ void resq_kernel(const float* __restrict__ W, const int* __restrict__ idx,
                            float* __restrict__ res, float* __restrict__ qout, int lvl) {
  size_t i = (size_t)blockIdx.x * blockDim.x + threadIdx.x;
  if (i >= (size_t)N_PTS * EDIM) return;
  int b = (int)(i >> 7), t = (int)(i & 127);
  float emb = W[(size_t)idx[b] * EDIM + t];
  qout[i] = (lvl == 0) ? emb : (qout[i] + emb);
  res[i] -= emb;
}

// ===========================================================================
extern "C" void kernel_launch(void* const* d_in, const int* in_sizes, int n_in,
                              void* d_out, int out_size, void* d_ws, size_t ws_size,
                              hipStream_t stream) {
  const float* inp   = (const float*)d_in[0];   // (N,128)
  const float* Wts   = (const float*)d_in[1];   // (4,4097,128)
  const float* csema = (const float*)d_in[2];   // (4,4096)
  const float* eema  = (const float*)d_in[3];   // (4,4096,128)
  float* out = (float*)d_out;
  float* F = (float*)d_ws;

  float* xsq  = F + WS_XSQ;
  float* csq  = F + WS_CSQ;
  float* u    = F + WS_U;
  float* logu = F + WS_LOGU;
  float* S    = F + WS_S;
  float* cs   = F + WS_CS;
  float* csn  = F + WS_CSN;
  float* v    = F + WS_V;
  int*   idxp = (int*)(F + WS_IDX);
  float* vs   = F + WS_VS;
  float* res  = F + WS_RES;
  __hip_bfloat16* E = (__hip_bfloat16*)(F + WS_E);
  float* Dm   = F + WS_D;

  const size_t ND = (size_t)N_PTS * EDIM;
  const int ldsBytes = 2 * TILE * LDSP * sizeof(float);

  // residual <- inputs
  copy_f_kernel<<<4096, 256, 0, stream>>>(inp, res, ND);

  for (int lvl = 0; lvl < DEPTH; ++lvl) {
    const float* W = Wts + (size_t)lvl * (K_CODES + 1) * EDIM;

    // row norms
    rownorm_kernel<<<N_PTS / 8, 256, 0, stream>>>(res, xsq, N_PTS);
    rownorm_kernel<<<K_CODES / 8, 256, 0, stream>>>(W, csq, K_CODES);

    // stats init: sum=sumsq=0, minkey=all-ones
    fill_f_kernel<<<1, 32, 0, stream>>>(F, 0.0f, 2);
    fill_u_kernel<<<1, 32, 0, stream>>>((uint32_t*)(F + 2), 0xFFFFFFFFu, 1);

    // distance matrix (WMMA + TDM)
    gemm_dist_kernel<<<dim3(K_CODES / TILE, N_PTS / TILE), 512, ldsBytes, stream>>>(
        res, W, xsq, csq, Dm, F);

    finalize_stats_kernel<<<1, 1, 0, stream>>>(F);

    // E = exp(s*d + o) (bf16) + total sum -> g
    fill_f_kernel<<<1, 32, 0, stream>>>(F + 5, 0.0f, 1);
    build_E_kernel<<<8192, 256, 0, stream>>>(Dm, E, F, F + 5);
    finalize_g_kernel<<<1, 1, 0, stream>>>(F);

    // sinkhorn scale vectors
    fill_f_kernel<<<16, 256, 0, stream>>>(u, 1.0f, K_CODES);
    fill_f_kernel<<<256, 256, 0, stream>>>(v, 1.0f, N_PTS);

    for (int it = 0; it < 5; ++it) {
      fill_f_kernel<<<16, 256, 0, stream>>>(S, 0.0f, K_CODES);
      rowsumS_kernel<<<dim3(K_CODES / 256, N_PTS / 256), 256, 0, stream>>>(E, v, S);
      update_u_kernel<<<K_CODES / 256, 256, 0, stream>>>(u, logu, S, F);
      if (it < 4)  // final column normalization cannot change the argmax
        colsum_v_kernel<<<N_PTS, 256, 0, stream>>>(E, u, v, F);
    }

    argmax_kernel<<<N_PTS, 256, 0, stream>>>(Dm, logu, F, idxp, out + OUT_CODES, lvl);

    // segment sums
    fill_f_kernel<<<16, 256, 0, stream>>>(cs, 0.0f, K_CODES);
    fill_f_kernel<<<2048, 256, 0, stream>>>(vs, 0.0f, (size_t)K_CODES * EDIM);
    seg_kernel<<<N_PTS, 128, 0, stream>>>(idxp, res, cs, vs);

    // EMA + weight update
    fill_f_kernel<<<1, 32, 0, stream>>>(F + 7, 0.0f, 1);
    ema_cs_kernel<<<K_CODES / 256, 256, 0, stream>>>(
        csema + (size_t)lvl * K_CODES, cs, csn, out + OUT_NCS + (size_t)lvl * K_CODES, F + 7);
    upW_kernel<<<(K_CODES * EDIM) / 256, 256, 0, stream>>>(
        eema + (size_t)lvl * K_CODES * EDIM, vs, csn, F,
        out + OUT_NE + (size_t)lvl * K_CODES * EDIM,
        out + OUT_NW + (size_t)lvl * (K_CODES + 1) * EDIM);
    lastrow_kernel<<<1, 128, 0, stream>>>(W, out + OUT_NW + (size_t)lvl * (K_CODES + 1) * EDIM);

    // residual / quant update (uses ORIGINAL W per reference)
    resq_kernel<<<(unsigned)(ND / 256), 256, 0, stream>>>(W, idxp, res, out + OUT_Q, lvl);
  }
}